// GraphConv_14886356648681
// MI455X (gfx1250) — compile-verified
//
#include <hip/hip_runtime.h>

// Problem constants (match the reference harness).
#define N_USERS 100000
#define N_ITEMS 200000
#define N_TOTAL 300000   // N_USERS + N_ITEMS, divisible by 16
#define N_EDGES 6000000
#define DIM     64

typedef __attribute__((ext_vector_type(2))) float v2f;   // A/B frag: 16x4 f32 = 64 vals / 32 lanes
typedef __attribute__((ext_vector_type(8))) float v8f;   // C/D frag: 16x16 f32 = 256 vals / 32 lanes

__device__ __forceinline__ const float* node_row(const float* __restrict__ uf,
                                                 const float* __restrict__ itf,
                                                 int n) {
  return (n < N_USERS) ? (uf + (size_t)n * DIM)
                       : (itf + (size_t)(n - N_USERS) * DIM);
}

// -----------------------------------------------------------------------------
// Kernel 1: trans[n][d] = sum_k feat[n][k] * W[d][k] + b[d]   (x @ W^T + b)
// One wave = one 16x16 output tile, chained V_WMMA_F32_16X16X4_F32 over K=64.
// 75000 tiles total = 9375 blocks * 8 waves (exact cover -> EXEC all ones).
// -----------------------------------------------------------------------------
__global__ void __launch_bounds__(256) gemm_wmma_f32(
    const float* __restrict__ uf, const float* __restrict__ itf,
    const float* __restrict__ W,  const float* __restrict__ bias,
    float* __restrict__ trans) {
  const int lane = threadIdx.x & 31;
  const int wave = threadIdx.x >> 5;
  const int tile = blockIdx.x * 8 + wave;     // 0 .. 74999
  const int rowTile = tile >> 2;              // 0 .. 18749
  const int colTile = tile & 3;               // 0 .. 3

  const int half = lane >> 4;                 // 0 = lanes 0-15, 1 = lanes 16-31
  const int l16  = lane & 15;

  // A fragment source: row = M index (per lane), K split across lane halves.
  const int arow = rowTile * 16 + l16;
  const float* __restrict__ aptr = node_row(uf, itf, arow);
  // B fragment source: N index per lane; B[k][n] = W[n][k].
  const int ncol = colTile * 16 + l16;
  const float* __restrict__ wptr = W + (size_t)ncol * DIM;

  v8f acc = {};
#pragma unroll
  for (int kb = 0; kb < 16; ++kb) {
    const int k = kb * 4 + half * 2;          // lanes 0-15: K={0,1}; 16-31: K={2,3}
    v2f a, b;
    a.x = aptr[k];
    a.y = aptr[k + 1];
    b.x = wptr[k];
    b.y = wptr[k + 1];
    // 8 args: (neg_a, A, neg_b, B, c_mod, C, reuse_a, reuse_b)
    acc = __builtin_amdgcn_wmma_f32_16x16x4_f32(false, a, false, b,
                                                (short)0, acc, false, false);
  }

  // D layout: VGPR r -> M = r (lanes 0-15) / r+8 (lanes 16-31), N = lane&15.
  const float bv = bias[ncol];
#pragma unroll
  for (int r = 0; r < 8; ++r) {
    const int m = rowTile * 16 + r + half * 8;
    trans[(size_t)m * DIM + ncol] = acc[r] + bv;
  }
}

// -----------------------------------------------------------------------------
// Kernel 2: out = all_feat (identity residual). float4 grid-stride copy.
// -----------------------------------------------------------------------------
__global__ void __launch_bounds__(256) init_out_residual(
    const float* __restrict__ uf, const float* __restrict__ itf,
    float* __restrict__ out) {
  const size_t total  = (size_t)N_TOTAL * DIM;     // 19.2M floats
  const size_t ucut   = (size_t)N_USERS * DIM;     // 6.4M  (divisible by 4)
  const size_t stride = (size_t)gridDim.x * blockDim.x * 4;
  for (size_t j = ((size_t)blockIdx.x * blockDim.x + threadIdx.x) * 4;
       j < total; j += stride) {
    const float4 v = (j < ucut) ? *(const float4*)(uf + j)
                                : *(const float4*)(itf + (j - ucut));
    *(float4*)(out + j) = v;
  }
}

// -----------------------------------------------------------------------------
// Kernel 3: for each edge e: out[A_row[e], :] += A_val[e] * trans[A_col[e], :]
// 64 consecutive threads (2 waves) own one edge -> coalesced 256B gather and
// 64 coalesced global_atomic_add_f32 per edge. trans + out both fit in L2.
// -----------------------------------------------------------------------------
__global__ void __launch_bounds__(256) spmm_scatter(
    const int* __restrict__ Arow, const int* __restrict__ Acol,
    const float* __restrict__ Aval, const float* __restrict__ trans,
    float* __restrict__ out) {
  const int d   = threadIdx.x & 63;    // feature dim
  const int sub = threadIdx.x >> 6;    // 4 edges per 256-thread block pass
  const long estride = (long)gridDim.x * 4;
  for (long e = (long)blockIdx.x * 4 + sub; e < N_EDGES; e += estride) {
    const int   r = Arow[e];
    const int   c = Acol[e];
    const float v = Aval[e];
    atomicAdd(&out[(size_t)r * DIM + d], v * trans[(size_t)c * DIM + d]);
  }
}

// -----------------------------------------------------------------------------
// Host-side launch. Inputs (setup_inputs order):
//   0: A_row int32[6M], 1: A_col int32[6M], 2: A_val f32[6M],
//   3: user_feat f32[100000*64], 4: item_feat f32[200000*64],
//   5: W f32[64*64], 6: b f32[64]
// d_out: f32[300000*64] (users then items, flat). d_ws: >= 76.8 MB for trans.
// -----------------------------------------------------------------------------
extern "C" void kernel_launch(void* const* d_in, const int* in_sizes, int n_in,
                              void* d_out, int out_size, void* d_ws, size_t ws_size,
                              hipStream_t stream) {
  (void)in_sizes; (void)n_in; (void)out_size; (void)ws_size;
  const int*   A_row = (const int*)d_in[0];
  const int*   A_col = (const int*)d_in[1];
  const float* A_val = (const float*)d_in[2];
  const float* uf    = (const float*)d_in[3];
  const float* itf   = (const float*)d_in[4];
  const float* W     = (const float*)d_in[5];
  const float* bias  = (const float*)d_in[6];
  float* out   = (float*)d_out;
  float* trans = (float*)d_ws;   // 300000 * 64 * 4 B = 76.8 MB scratch

  gemm_wmma_f32<<<9375, 256, 0, stream>>>(uf, itf, W, bias, trans);
  init_out_residual<<<2048, 256, 0, stream>>>(uf, itf, out);
  spmm_scatter<<<12288, 256, 0, stream>>>(A_row, A_col, A_val, trans, out);
}